// Rollout_7009386627075
// MI455X (gfx1250) — compile-verified
//
#include <hip/hip_runtime.h>
#include <math.h>

typedef float v2f __attribute__((ext_vector_type(2)));
typedef float v4f __attribute__((ext_vector_type(4)));
typedef float v8f __attribute__((ext_vector_type(8)));

#define GAMMA 0.99f
#define COEF  (0.99f * 0.95f)          // gamma * lambda
#define LOG2C (-0.08851485f)           // log2(0.9405)

static constexpr int Bdim = 4096;
static constexpr int Tdim = 8192;
static constexpr int NBUF = 4;         // async pipeline depth

// One wave handles 16 batch rows, marching backwards over T in chunks of 16
// timesteps. Per chunk: A_tile = W x Delta via 4x v_wmma_f32_16x16x4_f32
// (K=16), carry folded into Delta[K=15]. Input tiles are staged with
// depth-4 double... quad-buffered global_load_async_to_lds_b128 (ASYNCcnt),
// outputs streamed with non-temporal b128 stores after an LDS transpose.
__global__ void __launch_bounds__(32)
gae_wmma_kernel(const float* __restrict__ rewards,
                const float* __restrict__ values,
                float* __restrict__ adv,
                float* __restrict__ tgt) {
    __shared__ __align__(16) float s_r[NBUF][16][20];  // rewards tiles [buf][b][k]
    __shared__ __align__(16) float s_v[NBUF][16][20];  // values tiles, k=0..16 used
    __shared__ __align__(16) float s_a[16][20];        // advantages out tile
    __shared__ __align__(16) float s_t[16][20];        // targets out tile

    const int lane = threadIdx.x;      // 0..31
    const int half = lane >> 4;        // 0 / 1
    const int m    = lane & 15;        // column (batch row within group) in B/C layout
    const int row0 = blockIdx.x * 16;  // first batch row of this group

    // coalesced lane mapping: lane -> (row lrow / lrow+8, 4 consecutive t)
    const int lrow = lane >> 2;        // 0..7
    const int lcol = (lane & 3) * 4;   // 0,4,8,12

    const size_t vstride = (size_t)(Tdim + 1);

    // ---- constant A-matrix fragments: W[i][k] = c^(k-i) if k>=i else 0 ----
    v2f aw[4];
#pragma unroll
    for (int s4 = 0; s4 < 4; ++s4) {
#pragma unroll
        for (int v = 0; v < 2; ++v) {
            int kg = 4 * s4 + 2 * half + v;           // global K in 0..15
            float p = exp2f((float)(kg - m) * LOG2C); // c^(kg-m)
            aw[s4][v] = (kg >= m) ? p : 0.0f;
        }
    }

    const int i0 = 8 * half;           // C/D layout: VGPR r -> M = r + 8*half

    // ---- LDS byte offsets (flat ptr low 32 bits == LDS offset) for async dst
    uint32_t ldsR0[NBUF], ldsR1[NBUF], ldsV0[NBUF], ldsV1[NBUF];
#pragma unroll
    for (int b = 0; b < NBUF; ++b) {
        ldsR0[b] = (uint32_t)(uintptr_t)&s_r[b][lrow][lcol];
        ldsR1[b] = (uint32_t)(uintptr_t)&s_r[b][lrow + 8][lcol];
        ldsV0[b] = (uint32_t)(uintptr_t)&s_v[b][lrow][lcol];
        ldsV1[b] = (uint32_t)(uintptr_t)&s_v[b][lrow + 8][lcol];
    }
    const uint64_t rbase = (uint64_t)(uintptr_t)rewards;
    const uint64_t vbase = (uint64_t)(uintptr_t)values;

    // 4 async b128 loads stage one chunk (2KB) straight into LDS
    auto issue_async = [&](int s, int buf) {
        uint32_t ro0 = (uint32_t)(((row0 + lrow) * Tdim + s + lcol) * 4);
        uint32_t ro1 = ro0 + (uint32_t)(8 * Tdim * 4);
        uint32_t vo0 = (uint32_t)(((row0 + lrow) * (Tdim + 1) + s + lcol) * 4);
        uint32_t vo1 = vo0 + (uint32_t)(8 * (Tdim + 1) * 4);
        asm volatile(
            "global_load_async_to_lds_b128 %0, %4, %8\n\t"
            "global_load_async_to_lds_b128 %1, %5, %8\n\t"
            "global_load_async_to_lds_b128 %2, %6, %9\n\t"
            "global_load_async_to_lds_b128 %3, %7, %9"
            :: "v"(ldsR0[buf]), "v"(ldsR1[buf]), "v"(ldsV0[buf]), "v"(ldsV1[buf]),
               "v"(ro0), "v"(ro1), "v"(vo0), "v"(vo1),
               "s"(rbase), "s"(vbase)
            : "memory");
    };

    // prologue: fill all four buffers (prefetch 4 chunks deep)
#pragma unroll
    for (int b = 0; b < NBUF; ++b) issue_async(Tdim - 16 * (b + 1), b);

    float cy = 0.0f;                                             // carry A[b][s+16]
    float vbound = values[(size_t)(row0 + m) * vstride + Tdim];  // V[b][T]

    int it = 0;
    for (int s = Tdim - 16; s >= 0; s -= 16, ++it) {
        const int buf = it & (NBUF - 1);

        // Async loads complete in order: wait until only the NEWER chunks'
        // loads (4 each) remain in flight -> current buffer has landed.
        if      (s >= 48) asm volatile("s_wait_asynccnt 12" ::: "memory");
        else if (s == 32) asm volatile("s_wait_asynccnt 8"  ::: "memory");
        else if (s == 16) asm volatile("s_wait_asynccnt 4"  ::: "memory");
        else              asm volatile("s_wait_asynccnt 0"  ::: "memory");

        if (half == 0) s_v[buf][m][16] = vbound;  // boundary column V[b][s+16]

        // ---- B fragments: Delta[K][b], K = 4*s4 + 2*half + v ----
        v2f bf[4];
#pragma unroll
        for (int s4 = 0; s4 < 4; ++s4) {
            int kb = 4 * s4 + 2 * half;
            float r0 = s_r[buf][m][kb],  r1 = s_r[buf][m][kb + 1];
            float v0 = s_v[buf][m][kb],  v1 = s_v[buf][m][kb + 1];
            float v2 = s_v[buf][m][kb + 2];
            bf[s4][0] = r0 + GAMMA * v1 - v0;
            bf[s4][1] = r1 + GAMMA * v2 - v1;
        }
        // fold carry: Delta[15][b] += c * A[b][s+16] (K=15 at half=1,s4=3,v=1)
        if (half) bf[3][1] += COEF * cy;

        // ---- A_tile = W x Delta: two independent 2-deep WMMA chains ----
        v8f zero = {0.f, 0.f, 0.f, 0.f, 0.f, 0.f, 0.f, 0.f};
        v8f accH = __builtin_amdgcn_wmma_f32_16x16x4_f32(
            false, aw[3], false, bf[3], (short)0, zero, false, false);
        accH = __builtin_amdgcn_wmma_f32_16x16x4_f32(
            false, aw[2], false, bf[2], (short)0, accH, false, false);
        v8f accL = __builtin_amdgcn_wmma_f32_16x16x4_f32(
            false, aw[1], false, bf[1], (short)0, zero, false, false);
        accL = __builtin_amdgcn_wmma_f32_16x16x4_f32(
            false, aw[0], false, bf[0], (short)0, accL, false, false);

        float a[8];
#pragma unroll
        for (int r = 0; r < 8; ++r) a[r] = accL[r] + accH[r];

        // ---- targets: V[b][s+M] + A (C-layout read of V from LDS) ----
        v4f va0 = *(const v4f*)&s_v[buf][m][i0];
        v4f va1 = *(const v4f*)&s_v[buf][m][i0 + 4];
        float tg[8];
        tg[0] = va0.x + a[0]; tg[1] = va0.y + a[1];
        tg[2] = va0.z + a[2]; tg[3] = va0.w + a[3];
        tg[4] = va1.x + a[4]; tg[5] = va1.y + a[5];
        tg[6] = va1.z + a[6]; tg[7] = va1.w + a[7];

        // new carry = A[M=0][b] (a[0] of lanes 0..15), broadcast to both halves
        cy = __shfl(a[0], m, 32);

        // ---- C layout -> LDS row-major, then coalesced NT global stores ----
        *(v4f*)&s_a[m][i0]     = (v4f){a[0], a[1], a[2], a[3]};
        *(v4f*)&s_a[m][i0 + 4] = (v4f){a[4], a[5], a[6], a[7]};
        *(v4f*)&s_t[m][i0]     = (v4f){tg[0], tg[1], tg[2], tg[3]};
        *(v4f*)&s_t[m][i0 + 4] = (v4f){tg[4], tg[5], tg[6], tg[7]};

        v4f oa0 = *(const v4f*)&s_a[lrow][lcol];
        v4f oa1 = *(const v4f*)&s_a[lrow + 8][lcol];
        v4f ot0 = *(const v4f*)&s_t[lrow][lcol];
        v4f ot1 = *(const v4f*)&s_t[lrow + 8][lcol];
        __builtin_nontemporal_store(oa0, (v4f*)(adv + (size_t)(row0 + lrow)     * Tdim + s + lcol));
        __builtin_nontemporal_store(oa1, (v4f*)(adv + (size_t)(row0 + lrow + 8) * Tdim + s + lcol));
        __builtin_nontemporal_store(ot0, (v4f*)(tgt + (size_t)(row0 + lrow)     * Tdim + s + lcol));
        __builtin_nontemporal_store(ot1, (v4f*)(tgt + (size_t)(row0 + lrow + 8) * Tdim + s + lcol));

        // next chunk's boundary V[b][s] is this chunk's column 0 (read before
        // the buffer is re-armed below)
        vbound = s_v[buf][m][0];

        // all LDS reads of `buf` must retire before the async overwrite lands
        asm volatile("s_wait_dscnt 0" ::: "memory");
        if (s >= 16 * NBUF) issue_async(s - 16 * NBUF, buf);
    }
}

extern "C" void kernel_launch(void* const* d_in, const int* in_sizes, int n_in,
                              void* d_out, int out_size, void* d_ws, size_t ws_size,
                              hipStream_t stream) {
    (void)in_sizes; (void)n_in; (void)out_size; (void)d_ws; (void)ws_size;
    const float* rewards = (const float*)d_in[0];          // [B, T]
    const float* values  = (const float*)d_in[1];          // [B, T+1]
    float* adv = (float*)d_out;                            // [B, T]
    float* tgt = adv + (size_t)Bdim * Tdim;                // [B, T]
    dim3 grid(Bdim / 16), block(32);
    hipLaunchKernelGGL(gae_wmma_kernel, grid, block, 0, stream,
                       rewards, values, adv, tgt);
}